// PolarityPairMixer_8675833938103
// MI455X (gfx1250) — compile-verified
//
#include <hip/hip_runtime.h>
#include <math.h>

#define B_ 4
#define C_ 256
#define V_ 512
#define D_ 128

typedef __attribute__((ext_vector_type(2))) float v2f;
typedef __attribute__((ext_vector_type(4))) float v4f;
typedef __attribute__((ext_vector_type(8))) float v8f;

__device__ __forceinline__ float wave_sum32(float x) {
#pragma unroll
  for (int off = 16; off > 0; off >>= 1) x += __shfl_xor(x, off, 32);
  return x;
}

__device__ __forceinline__ float gelu_exact(float x) {
  return 0.5f * x * (1.0f + erff(x * 0.70710678118654752f));
}

// ---------------------------------------------------------------------------
// K1: masked mean-pool over C, build pair_feat = [p, n, p-n, p*n], LN(512).
// One wave (32 lanes) per (b,v) row; each lane owns 4 of the 128 channels.
// ---------------------------------------------------------------------------
__global__ void pool_feat_ln_kernel(const float* __restrict__ cell,
                                    const float* __restrict__ vsm,
                                    const float* __restrict__ pn_g,
                                    const float* __restrict__ pn_b,
                                    float* __restrict__ feat) {
  const int lane = threadIdx.x & 31;
  const int wave = threadIdx.x >> 5;
  const int bv   = blockIdx.x * 8 + wave;   // 2048 rows total
  const int b    = bv >> 9;                 // V_ == 512
  const int v    = bv & 511;

  const float* cbase = cell + ((size_t)b * C_ * V_ + (size_t)v) * D_ + lane * 4;
  const float* pmp   = vsm + ((size_t)(b * 2 + 0) * C_) * V_ + v;
  const float* nmp   = vsm + ((size_t)(b * 2 + 1) * C_) * V_ + v;

  v4f psum = {}; v4f nsum = {};
  float pc = 0.f, nc = 0.f;
  for (int c = 0; c < C_; ++c) {
    const float pm = (pmp[(size_t)c * V_] > 0.f) ? 1.f : 0.f;
    const float nm = (nmp[(size_t)c * V_] > 0.f) ? 1.f : 0.f;
    pc += pm; nc += nm;
    const v4f x = *(const v4f*)(cbase + (size_t)c * V_ * D_);
    psum += pm * x;
    nsum += nm * x;
  }

  const float pr = 1.f / fmaxf(pc, 1.f);
  const float nr = 1.f / fmaxf(nc, 1.f);
  const v4f p   = psum * pr;
  const v4f n   = nsum * nr;
  const v4f dmn = p - n;
  const v4f pmn = p * n;

  float ls = 0.f, ls2 = 0.f;
#pragma unroll
  for (int j = 0; j < 4; ++j) {
    ls  += p[j] + n[j] + dmn[j] + pmn[j];
    ls2 += p[j]*p[j] + n[j]*n[j] + dmn[j]*dmn[j] + pmn[j]*pmn[j];
  }
  const float s    = wave_sum32(ls);
  const float s2   = wave_sum32(ls2);
  const float mean = s * (1.f / 512.f);
  const float var  = s2 * (1.f / 512.f) - mean * mean;
  const float rs   = rsqrtf(var + 1e-5f);

  float* fo = feat + (size_t)bv * 512;
  const int d0 = lane * 4;
  v4f o0, o1, o2, o3;
#pragma unroll
  for (int j = 0; j < 4; ++j) {
    o0[j] = (p[j]   - mean) * rs * pn_g[d0 + j]       + pn_b[d0 + j];
    o1[j] = (n[j]   - mean) * rs * pn_g[128 + d0 + j] + pn_b[128 + d0 + j];
    o2[j] = (dmn[j] - mean) * rs * pn_g[256 + d0 + j] + pn_b[256 + d0 + j];
    o3[j] = (pmn[j] - mean) * rs * pn_g[384 + d0 + j] + pn_b[384 + d0 + j];
  }
  *(v4f*)(fo + d0)       = o0;
  *(v4f*)(fo + 128 + d0) = o1;
  *(v4f*)(fo + 256 + d0) = o2;
  *(v4f*)(fo + 384 + d0) = o3;
}

// ---------------------------------------------------------------------------
// Generic f32 WMMA GEMM: C[M,N] = act(A[M,K] @ W[K,N] + bias[N]).
// One 16x16 output tile per wave via V_WMMA_F32_16X16X4_F32.
// A fragment (16x4): lanes 0-15 M=lane, VGPR0 holds K={0 | 2}, VGPR1 K={1 | 3}
// (lane halves select K pair). B fragment mirrored (lane = N).
// D fragment: VGPR r -> M = r + 8*(lane>>4), N = lane&15.
// ---------------------------------------------------------------------------
__global__ void wmma_gemm_kernel(const float* __restrict__ A,
                                 const float* __restrict__ W,
                                 const float* __restrict__ bias,
                                 float* __restrict__ Cout,
                                 int M, int N, int K, int act) {
  const int lane = threadIdx.x & 31;
  const int wave = threadIdx.x >> 5;
  const int tile = blockIdx.x * 8 + wave;
  const int ntn  = N >> 4;
  const int tm   = tile / ntn;
  const int tn   = tile - tm * ntn;
  if (tm * 16 >= M) return;  // wave-uniform: EXEC stays all-ones

  const int l  = lane & 15;
  const int hi = lane >> 4;

  const int arow = tm * 16 + l;
  const int col  = tn * 16 + l;
  const float* Ap = A + (size_t)arow * K + 2 * hi;      // a0=Ap[k], a1=Ap[k+1]
  const float* Wp = W + (size_t)(2 * hi) * N + col;     // b0=Wp[k*N], b1=Wp[k*N+N]

  v8f acc = {};
  for (int k = 0; k < K; k += 4) {
    v2f a, bb;
    a[0]  = Ap[k];
    a[1]  = Ap[k + 1];
    bb[0] = Wp[(size_t)k * N];
    bb[1] = Wp[(size_t)k * N + N];
    acc = __builtin_amdgcn_wmma_f32_16x16x4_f32(
        /*neg_a=*/false, a, /*neg_b=*/false, bb,
        /*c_mod=*/(short)0, acc, /*reuse_a=*/false, /*reuse_b=*/false);
  }

  const float bv = bias[col];
#pragma unroll
  for (int r = 0; r < 8; ++r) {
    const int row = tm * 16 + 8 * hi + r;
    float x = acc[r] + bv;
    if (act == 1) x = gelu_exact(x);
    Cout[(size_t)row * N + col] = x;
  }
}

// ---------------------------------------------------------------------------
// K5: gate head. gs[row] = 0.1 * var_mask * sigmoid(g[row,:64] . gw2 + gb2)
// ---------------------------------------------------------------------------
__global__ void gate_kernel(const float* __restrict__ g,
                            const float* __restrict__ gw2,
                            const float* __restrict__ gb2,
                            const int* __restrict__ var_mask,
                            float* __restrict__ gs) {
  const int row = blockIdx.x * blockDim.x + threadIdx.x;
  if (row >= B_ * V_) return;
  float acc = gb2[0];
  const float* gr = g + (size_t)row * 64;
#pragma unroll 8
  for (int j = 0; j < 64; ++j) acc += gr[j] * gw2[j];
  const float gate = 1.f / (1.f + expf(-acc));
  const float m = (var_mask[row] != 0) ? 1.f : 0.f;
  gs[row] = 0.1f * gate * m;
}

// ---------------------------------------------------------------------------
// K6: out = LN_D( cell + sign * gs[b,v] * delta[b,v,:] ).
// One wave per (b,c,v) row of 128 channels; float4 per lane; shuffle-reduce LN.
// ---------------------------------------------------------------------------
__global__ void writeback_ln_kernel(const float* __restrict__ cell,
                                    const float* __restrict__ vsm,
                                    const float* __restrict__ delta,
                                    const float* __restrict__ gs,
                                    const float* __restrict__ on_g,
                                    const float* __restrict__ on_b,
                                    float* __restrict__ out) {
  const int lane = threadIdx.x & 31;
  const int wave = threadIdx.x >> 5;
  const int r    = blockIdx.x * 8 + wave;   // row in [0, B*C*V)
  const int b    = r >> 17;                 // C_*V_ == 131072
  const int rem  = r & 131071;
  const int c    = rem >> 9;
  const int v    = rem & 511;

  const float pm = vsm[((size_t)(b * 2 + 0) * C_ + c) * V_ + v];
  const float nm = vsm[((size_t)(b * 2 + 1) * C_ + c) * V_ + v];
  const float sgn = ((pm > 0.f) ? 1.f : 0.f) - ((nm > 0.f) ? 1.f : 0.f);
  const int bv = b * V_ + v;
  const float scale = sgn * gs[bv];

  const size_t off = (size_t)r * D_ + lane * 4;
  v4f x = *(const v4f*)(cell + off);
  const v4f d4 = *(const v4f*)(delta + (size_t)bv * D_ + lane * 4);
  x += scale * d4;

  float ls = x[0] + x[1] + x[2] + x[3];
  float ls2 = x[0]*x[0] + x[1]*x[1] + x[2]*x[2] + x[3]*x[3];
  const float s    = wave_sum32(ls);
  const float s2   = wave_sum32(ls2);
  const float mean = s * (1.f / 128.f);
  const float var  = s2 * (1.f / 128.f) - mean * mean;
  const float rs   = rsqrtf(var + 1e-5f);

  const int d0 = lane * 4;
  v4f y;
#pragma unroll
  for (int j = 0; j < 4; ++j)
    y[j] = (x[j] - mean) * rs * on_g[d0 + j] + on_b[d0 + j];
  *(v4f*)(out + off) = y;
}

// ---------------------------------------------------------------------------
extern "C" void kernel_launch(void* const* d_in, const int* in_sizes, int n_in,
                              void* d_out, int out_size, void* d_ws, size_t ws_size,
                              hipStream_t stream) {
  const float* cell     = (const float*)d_in[0];
  const float* vsm      = (const float*)d_in[1];
  const int*   var_mask = (const int*)d_in[2];
  const float* pn_g     = (const float*)d_in[3];
  const float* pn_b     = (const float*)d_in[4];
  const float* dw1      = (const float*)d_in[5];
  const float* db1      = (const float*)d_in[6];
  const float* dw2      = (const float*)d_in[7];
  const float* db2      = (const float*)d_in[8];
  const float* gw1      = (const float*)d_in[9];
  const float* gb1      = (const float*)d_in[10];
  const float* gw2      = (const float*)d_in[11];
  const float* gb2      = (const float*)d_in[12];
  const float* on_g     = (const float*)d_in[13];
  const float* on_b     = (const float*)d_in[14];
  float* out = (float*)d_out;

  // Workspace layout (floats): feat[2048*512] h[2048*256] g[2048*64]
  //                            delta[2048*128] gs[2048]  (~7.6 MB total)
  float* ws    = (float*)d_ws;
  float* feat  = ws;
  float* h     = feat  + 2048 * 512;
  float* g     = h     + 2048 * 256;
  float* delta = g     + 2048 * 64;
  float* gs    = delta + 2048 * 128;

  const int ROWS = B_ * V_;  // 2048

  // K1: pool + pair_feat + LN(512)          (reads 256 MB of cell_emb)
  pool_feat_ln_kernel<<<ROWS / 8, 256, 0, stream>>>(cell, vsm, pn_g, pn_b, feat);

  // K2: h = gelu(feat @ dw1 + db1)          2048x512x256
  wmma_gemm_kernel<<<(ROWS / 16) * (256 / 16) / 8, 256, 0, stream>>>(
      feat, dw1, db1, h, ROWS, 256, 512, 1);

  // K3: g = gelu(feat @ gw1 + gb1)          2048x512x64
  wmma_gemm_kernel<<<(ROWS / 16) * (64 / 16) / 8, 256, 0, stream>>>(
      feat, gw1, gb1, g, ROWS, 64, 512, 1);

  // K4: delta = h @ dw2 + db2               2048x256x128
  wmma_gemm_kernel<<<(ROWS / 16) * (128 / 16) / 8, 256, 0, stream>>>(
      h, dw2, db2, delta, ROWS, 128, 256, 0);

  // K5: gate scalar per (b,v), folds sigmoid, var_mask and writeback_scale
  gate_kernel<<<(ROWS + 255) / 256, 256, 0, stream>>>(g, gw2, gb2, var_mask, gs);

  // K6: writeback + LN(128)                 (reads 256 MB, writes 256 MB)
  writeback_ln_kernel<<<(B_ * C_ * V_) / 8, 256, 0, stream>>>(
      cell, vsm, delta, gs, on_g, on_b, out);
}